// Parser_90159953477943
// MI455X (gfx1250) — compile-verified
//
#include <hip/hip_runtime.h>
#include <hip/hip_bf16.h>

// ---------------------------------------------------------------------------
// PYSEQM parser on MI455X (gfx1250).
//
//  - L2-resident problem (2 MB inputs vs 192 MB L2): latency/LDS bound, not
//    HBM bound. The win is the per-molecule 128x128 distance screen on the
//    matrix pipe: d^2(a,b) = |a|^2+|b|^2 - 2*(C C^T)[a,b], C in R^{128x3},
//    tiled 16x16 with V_WMMA_F32_16X16X4_F32 (K=4, z-padded). Upper-triangle
//    tiles only (tj >= wave).
//  - Blank atoms encoded as nrm=+inf -> their d^2 screens to +inf, no flag[].
//  - Atom data in LDS as float4 (x,y,z,nrm) -> single ds_load_b128 per atom.
//  - Branchless predicate + wave32 ballot: one ballot per 16x16 tile row pair;
//    rows are wave-private so bitmask halfword stores need no atomics.
//  - Deterministic compaction everywhere (prefix sums); emit order is flat
//    (a,b) row-major, matching jnp.where. Data-dependent output offsets
//    (n_real, npairs) computed on-device -> graph-capture safe.
// ---------------------------------------------------------------------------

#define NMOL     1024
#define MOLSIZE  128
#define NATOM    (NMOL * MOLSIZE)   // 131072
#define ABLK     256
#define NABLK    (NATOM / ABLK)     // 512
#define CUTOFF2  25.0f
#define SCREEN2  25.5f              // loose screen on WMMA d^2 (fp slack)
#define LCONV    1.8897261258369282f

typedef float v2f __attribute__((ext_vector_type(2)));
typedef float v8f __attribute__((ext_vector_type(8)));

union V8 { v8f v; float f[8]; };

// Workspace layout (int32 indices into d_ws)
#define WS_NREAL  0
#define WS_NPAIR  1
#define WS_BLKCNT 16
#define WS_BLKOFF (WS_BLKCNT + NABLK)
#define WS_INV    (WS_BLKOFF + NABLK)       // inv_real_atoms[NATOM]
#define WS_PCNT   (WS_INV + NATOM)          // per-molecule pair counts
#define WS_PBASE  (WS_PCNT + NMOL)          // per-molecule pair bases

// Fixed d_out section offsets (floats); data-dependent sections use
// R = n_real, NP = npairs read from workspace on device.
#define OFF_NHEAVY 2
#define OFF_NHYDRO (2 + NMOL)
#define OFF_NOCC   (2 + 2 * NMOL)
#define OFF_Z      (2 + 3 * NMOL)           // 3074

// ---------------------------------------------------------------------------
// Pass A1: per-block non-blank counts.
// ---------------------------------------------------------------------------
__global__ void k_atom_block_count(const int* __restrict__ sp,
                                   int* __restrict__ ws) {
  __shared__ int red[ABLK];
  const int t = threadIdx.x;
  const int g = blockIdx.x * ABLK + t;
  red[t] = (sp[g] > 0) ? 1 : 0;
  __syncthreads();
  for (int s = ABLK / 2; s > 0; s >>= 1) {
    if (t < s) red[t] += red[t + s];
    __syncthreads();
  }
  if (t == 0) ws[WS_BLKCNT + blockIdx.x] = red[0];
}

// Pass A2: exclusive scan of 512 block counts; total -> ws[WS_NREAL].
__global__ void k_scan_blocks(int* __restrict__ ws) {
  __shared__ int tmp[NABLK];
  const int t = threadIdx.x;
  const int v0 = ws[WS_BLKCNT + t];
  tmp[t] = v0;
  __syncthreads();
  for (int off = 1; off < NABLK; off <<= 1) {
    int v = (t >= off) ? tmp[t - off] : 0;
    __syncthreads();
    tmp[t] += v;
    __syncthreads();
  }
  ws[WS_BLKOFF + t] = tmp[t] - v0;
  if (t == NABLK - 1) ws[WS_NREAL] = tmp[t];
}

// Pass A3: deterministic scatter of compacted atoms; also inv_real_atoms.
__global__ void k_atom_scatter(const int* __restrict__ sp,
                               int* __restrict__ ws,
                               float* __restrict__ out) {
  __shared__ int tmp[ABLK];
  const int t = threadIdx.x;
  const int g = blockIdx.x * ABLK + t;
  const int s = sp[g];
  const int f = (s > 0) ? 1 : 0;
  tmp[t] = f;
  __syncthreads();
  for (int off = 1; off < ABLK; off <<= 1) {
    int v = (t >= off) ? tmp[t - off] : 0;
    __syncthreads();
    tmp[t] += v;
    __syncthreads();
  }
  const int excl = tmp[t] - f;
  const int R = ws[WS_NREAL];
  if (f) {
    const int idx = ws[WS_BLKOFF + blockIdx.x] + excl;
    const int m = g >> 7;     // molecule
    const int a = g & 127;    // atom-in-molecule
    out[OFF_Z + idx]         = (float)s;                                   // Z
    out[OFF_Z + R + idx]     = (float)(m * MOLSIZE * MOLSIZE +
                                       a * (MOLSIZE + 1));                 // maskd
    out[OFF_Z + 2 * R + idx] = (float)m;                                   // atom_molid
    ws[WS_INV + g] = idx;
  } else {
    ws[WS_INV + g] = 0;
  }
}

// Per-molecule stats: nHeavy, nHydro, nocc (+ the two scalars).
__global__ void k_mol_stats(const int* __restrict__ sp,
                            const float* __restrict__ tore,
                            float* __restrict__ out) {
  __shared__ int   rh[MOLSIZE];
  __shared__ int   ry[MOLSIZE];
  __shared__ float rq[MOLSIZE];
  const int t = threadIdx.x;
  const int m = blockIdx.x;
  const int s = sp[m * MOLSIZE + t];
  rh[t] = (s > 1) ? 1 : 0;
  ry[t] = (s == 1) ? 1 : 0;
  rq[t] = (s >= 0 && s < 10) ? tore[s] : 0.0f;
  __syncthreads();
  for (int k = MOLSIZE / 2; k > 0; k >>= 1) {
    if (t < k) { rh[t] += rh[t + k]; ry[t] += ry[t + k]; rq[t] += rq[t + k]; }
    __syncthreads();
  }
  if (t == 0) {
    out[OFF_NHEAVY + m] = (float)rh[0];
    out[OFF_NHYDRO + m] = (float)ry[0];
    out[OFF_NOCC + m]   = (float)((int)(rq[0] * 0.5f));  // trunc like astype(int32)
    if (m == 0) { out[0] = (float)NMOL; out[1] = (float)MOLSIZE; }
  }
}

// ---------------------------------------------------------------------------
// Shared per-molecule setup for the pair kernels: LDS float4 (x,y,z,nrm),
// blanks marked with nrm = +inf so they fail the d^2 screen arithmetically.
// ---------------------------------------------------------------------------
__device__ __forceinline__ void load_mol(const int* __restrict__ sp,
                                         const float* __restrict__ coord,
                                         int m, int t, float4* c4) {
  if (t < MOLSIZE) {
    const float* c = coord + (size_t)(m * MOLSIZE + t) * 3;
    const float x = c[0], y = c[1], z = c[2];
    const float nrm = (sp[m * MOLSIZE + t] > 0) ? (x * x + y * y + z * z)
                                                : __builtin_inff();
    c4[t] = make_float4(x, y, z, nrm);
  }
}

// ---------------------------------------------------------------------------
// Pass B1: per-molecule pair count. One workgroup (8 waves) per molecule;
// wave w owns row-block w and column blocks tj >= w. Fully branchless screen;
// ballot+popcount counting.
// ---------------------------------------------------------------------------
__global__ void k_pair_count(const int* __restrict__ sp,
                             const float* __restrict__ coord,
                             int* __restrict__ ws) {
  __shared__ float4 c4[MOLSIZE];
  __shared__ int wred[8];
  const int t = threadIdx.x;
  const int m = blockIdx.x;
  load_mol(sp, coord, m, t, c4);
  __syncthreads();

  const int lane = t & 31, wave = t >> 5;
  const int half = lane >> 4, lm = lane & 15;
  const int base = wave * 16 + half * 8;

  // A-fragment (16x4, K = x,y | z,0) — invariant over tj.
  const float4 ca = c4[wave * 16 + lm];
  v2f A;
  A.x = half ? ca.z : ca.x;
  A.y = half ? 0.0f : ca.y;

  // a-side row data for this lane's 8 result rows — invariant over tj.
  float4 rowc[8];
#pragma unroll
  for (int r = 0; r < 8; ++r) rowc[r] = c4[base + r];

  int cnt = 0;
  for (int tj = wave; tj < 8; ++tj) {          // uniform per wave -> EXEC full
    const int b = tj * 16 + lm;
    const float4 cb = c4[b];
    v2f B;
    B.x = half ? cb.z : cb.x;
    B.y = half ? 0.0f : cb.y;
    v8f C = {0.f, 0.f, 0.f, 0.f, 0.f, 0.f, 0.f, 0.f};
    V8 D;
    D.v = __builtin_amdgcn_wmma_f32_16x16x4_f32(false, A, false, B,
                                                (short)0, C, false, false);
#pragma unroll
    for (int r = 0; r < 8; ++r) {
      const int a = base + r;
      const float d2 = rowc[r].w + cb.w - 2.0f * D.f[r];
      const float dx = cb.x - rowc[r].x;
      const float dy = cb.y - rowc[r].y;
      const float dz = cb.z - rowc[r].z;
      const float dd = dx * dx + dy * dy + dz * dz;
      const bool pass = (a < b) & (d2 < SCREEN2) & (dd < CUTOFF2);
      cnt += __builtin_popcount(__builtin_amdgcn_ballot_w32(pass));
    }
  }
  if (lane == 0) wred[wave] = cnt;   // cnt uniform across the wave
  __syncthreads();
  if (t == 0) {
    int s = 0;
#pragma unroll
    for (int w = 0; w < 8; ++w) s += wred[w];
    ws[WS_PCNT + m] = s;
  }
}

// Pass B2: exclusive scan of 1024 molecule pair counts; total -> ws[WS_NPAIR].
__global__ void k_scan_pairs(int* __restrict__ ws) {
  __shared__ int tmp[NMOL];
  const int t = threadIdx.x;
  const int v0 = ws[WS_PCNT + t];
  tmp[t] = v0;
  __syncthreads();
  for (int off = 1; off < NMOL; off <<= 1) {
    int v = (t >= off) ? tmp[t - off] : 0;
    __syncthreads();
    tmp[t] += v;
    __syncthreads();
  }
  ws[WS_PBASE + t] = tmp[t] - v0;
  if (t == NMOL - 1) ws[WS_NPAIR] = tmp[t];
}

// ---------------------------------------------------------------------------
// Pass B3: ordered emit. Same WMMA screen; each ballot yields two 16-column
// row segments stored as halfwords into a wave-private row bitmask (no
// atomics). Row-count + row-scan then assigns each pair its deterministic
// slot in flat (a,b) order, matching jnp.where.
// ---------------------------------------------------------------------------
__global__ void k_pair_emit(const int* __restrict__ sp,
                            const float* __restrict__ coord,
                            const int* __restrict__ ws,
                            float* __restrict__ out) {
  __shared__ float4 c4[MOLSIZE];
  __shared__ unsigned short bm16[MOLSIZE][8];   // row bitmask, 16 cols/halfword
  __shared__ int rowcnt[MOLSIZE], rowoff[MOLSIZE];
  const int t = threadIdx.x;
  const int m = blockIdx.x;
  load_mol(sp, coord, m, t, c4);
  if (t < MOLSIZE) {
    unsigned* z = (unsigned*)&bm16[t][0];
    z[0] = 0u; z[1] = 0u; z[2] = 0u; z[3] = 0u;
  }
  __syncthreads();

  const int lane = t & 31, wave = t >> 5;
  const int half = lane >> 4, lm = lane & 15;
  const int base = wave * 16 + half * 8;

  const float4 ca = c4[wave * 16 + lm];
  v2f A;
  A.x = half ? ca.z : ca.x;
  A.y = half ? 0.0f : ca.y;

  float4 rowc[8];
#pragma unroll
  for (int r = 0; r < 8; ++r) rowc[r] = c4[base + r];

  for (int tj = wave; tj < 8; ++tj) {
    const int b = tj * 16 + lm;
    const float4 cb = c4[b];
    v2f B;
    B.x = half ? cb.z : cb.x;
    B.y = half ? 0.0f : cb.y;
    v8f C = {0.f, 0.f, 0.f, 0.f, 0.f, 0.f, 0.f, 0.f};
    V8 D;
    D.v = __builtin_amdgcn_wmma_f32_16x16x4_f32(false, A, false, B,
                                                (short)0, C, false, false);
#pragma unroll
    for (int r = 0; r < 8; ++r) {
      const int a = base + r;
      const float d2 = rowc[r].w + cb.w - 2.0f * D.f[r];
      const float dx = cb.x - rowc[r].x;
      const float dy = cb.y - rowc[r].y;
      const float dz = cb.z - rowc[r].z;
      const float dd = dx * dx + dy * dy + dz * dz;
      const bool pass = (a < b) & (d2 < SCREEN2) & (dd < CUTOFF2);
      const unsigned bal = __builtin_amdgcn_ballot_w32(pass);
      if (lm == 0)  // lanes 0 and 16: one 16-col segment each, wave-private row
        bm16[base + r][tj] =
            (unsigned short)(half ? (bal >> 16) : (bal & 0xffffu));
    }
  }
  __syncthreads();

  if (t < MOLSIZE) {
    const unsigned* w = (const unsigned*)&bm16[t][0];
    rowcnt[t] = __builtin_popcount(w[0]) + __builtin_popcount(w[1]) +
                __builtin_popcount(w[2]) + __builtin_popcount(w[3]);
    rowoff[t] = rowcnt[t];
  }
  __syncthreads();
  for (int off = 1; off < MOLSIZE; off <<= 1) {
    int v = 0;
    if (t < MOLSIZE && t >= off) v = rowoff[t - off];
    __syncthreads();
    if (t < MOLSIZE) rowoff[t] += v;
    __syncthreads();
  }

  const int R  = ws[WS_NREAL];
  const int NP = ws[WS_NPAIR];
  const long P0 = (long)OFF_Z + 3L * R;
  if (t < MOLSIZE) {
    const int a = t;
    long p = (long)ws[WS_PBASE + m] + (rowoff[a] - rowcnt[a]);
    const int flatA = m * MOLSIZE + a;
    const float4 caa = c4[a];
    const unsigned* wv = (const unsigned*)&bm16[a][0];
#pragma unroll
    for (int w = 0; w < 4; ++w) {
      unsigned bits = wv[w];
      while (bits) {
        const int bit = __builtin_ctz(bits);
        bits &= bits - 1;
        const int b = w * 32 + bit;
        const int flatB = m * MOLSIZE + b;
        const float4 cbb = c4[b];
        const float dx = cbb.x - caa.x, dy = cbb.y - caa.y, dz = cbb.z - caa.z;
        const float dist = __builtin_amdgcn_sqrtf(dx * dx + dy * dy + dz * dz);
        const float inv = __builtin_amdgcn_rcpf(dist);
        out[P0 + p]            = (float)(flatA * MOLSIZE + b);      // mask (<2^24)
        out[P0 + 1L * NP + p]  = (float)m;                          // pair_molid
        out[P0 + 2L * NP + p]  = (float)sp[flatA];                  // ni
        out[P0 + 3L * NP + p]  = (float)sp[flatB];                  // nj
        out[P0 + 4L * NP + p]  = (float)ws[WS_INV + flatA];         // idxi
        out[P0 + 5L * NP + p]  = (float)ws[WS_INV + flatB];         // idxj
        const long xo = P0 + 6L * NP + 3L * p;
        out[xo + 0] = dx * inv;                                     // xij
        out[xo + 1] = dy * inv;
        out[xo + 2] = dz * inv;
        out[P0 + 9L * NP + p]  = dist * LCONV;                      // rij
        ++p;
      }
    }
  }
}

extern "C" void kernel_launch(void* const* d_in, const int* in_sizes, int n_in,
                              void* d_out, int out_size, void* d_ws, size_t ws_size,
                              hipStream_t stream) {
  const int*   sp    = (const int*)d_in[0];    // species  (1024*128) int32
  const float* coord = (const float*)d_in[1];  // coords   (1024*128*3) f32
  const float* tore  = (const float*)d_in[2];  // tore     (10) f32
  float* out = (float*)d_out;
  int*   ws  = (int*)d_ws;

  k_atom_block_count<<<NABLK, ABLK, 0, stream>>>(sp, ws);
  k_scan_blocks     <<<1, NABLK, 0, stream>>>(ws);
  k_atom_scatter    <<<NABLK, ABLK, 0, stream>>>(sp, ws, out);
  k_mol_stats       <<<NMOL, MOLSIZE, 0, stream>>>(sp, tore, out);
  k_pair_count      <<<NMOL, 256, 0, stream>>>(sp, coord, ws);
  k_scan_pairs      <<<1, NMOL, 0, stream>>>(ws);
  k_pair_emit       <<<NMOL, 256, 0, stream>>>(sp, coord, ws, out);
}